// Correlation_85134841741354
// MI455X (gfx1250) — compile-verified
//
#include <hip/hip_runtime.h>
#include <hip/hip_bf16.h>

typedef __attribute__((ext_vector_type(2))) float v2f;
typedef __attribute__((ext_vector_type(8))) float v8f;

#define A_ATOMS 32
#define N_FEAT  128
#define SROW    132   // padded LDS row stride (floats) for staged S: conflict-free v2f reads
#define CROW    33    // padded row stride (floats) for 32x32 Gram matrix C

__device__ __forceinline__ float wsum32(float v) {
#pragma unroll
  for (int o = 16; o > 0; o >>= 1) v += __shfl_xor(v, o, 32);
  return v;
}
__device__ __forceinline__ float wmin32(float v) {
#pragma unroll
  for (int o = 16; o > 0; o >>= 1) v = fminf(v, __shfl_xor(v, o, 32));
  return v;
}

__global__ __launch_bounds__(32)
void Correlation_85134841741354_kernel(const float* __restrict__ sr,
                                       float* __restrict__ out, int M)
{
  __shared__ __align__(16) float lds[A_ATOMS * SROW];  // S staging; later aliased as C (stride CROW)
  __shared__ float vbuf[32], wbuf[32], dbuf[32], ebuf[32];

  const int lane = threadIdx.x;
  const int mol  = blockIdx.x;
  if (mol >= M) return;

  // ---------------- stage S (32x128 f32 = 16KB) into LDS ----------------
  // CDNA5 async copy global->LDS: no VGPR staging, tracked by ASYNCcnt.
  // Each lane moves 16B/row; per-lane LDS dest VGPR preserves the 132-float pad.
  const char* gbase = (const char*)(sr + (size_t)mol * (A_ATOMS * N_FEAT));
#pragma unroll
  for (int t = 0; t < A_ATOMS; ++t) {
    unsigned lds32 = (unsigned)(size_t)&lds[t * SROW + lane * 4];   // LDS byte offset (low 32 bits)
    const void* gp = gbase + ((size_t)t * N_FEAT + lane * 4) * sizeof(float);
    asm volatile("global_load_async_to_lds_b128 %0, %1, off"
                 :: "v"(lds32), "v"(gp) : "memory");
  }
  asm volatile("s_wait_asynccnt 0x0" ::: "memory");
  __syncthreads();

  // ---------------- Gram C = S*S^T via V_WMMA_F32_16X16X4_F32 ----------------
  // f32 16x4 A layout: lane m(<16): {K=k0,k0+1}; lane m+16: {K=k0+2,k0+3}.
  // For C = S*S^T the matching 4x16 B operand has the IDENTICAL per-lane data,
  // so one float2 per row-block per lane serves as both A and B.
  v8f acc00 = {}; v8f acc11 = {}; v8f acc01 = {};
  const int r    = lane & 15;
  const int koff = (lane >> 4) * 2;
#pragma unroll
  for (int k0 = 0; k0 < N_FEAT; k0 += 4) {
    v2f P = *(const v2f*)&lds[r        * SROW + k0 + koff];  // atom rows 0..15
    v2f Q = *(const v2f*)&lds[(16 + r) * SROW + k0 + koff];  // atom rows 16..31
    acc00 = __builtin_amdgcn_wmma_f32_16x16x4_f32(false, P, false, P, (short)0, acc00, false, false);
    acc11 = __builtin_amdgcn_wmma_f32_16x16x4_f32(false, Q, false, Q, (short)0, acc11, false, false);
    acc01 = __builtin_amdgcn_wmma_f32_16x16x4_f32(false, P, false, Q, (short)0, acc01, false, false);
  }
  __syncthreads();   // done with staged S; reuse LDS as C

  float* C = lds;
  const int tr = (lane >> 4) * 8;   // C/D layout: lanes 16-31 hold rows +8
  const int tc = lane & 15;
#pragma unroll
  for (int rr = 0; rr < 8; ++rr) {
    int row = rr + tr;
    C[row * CROW + tc]               = acc00[rr];
    C[(16 + row) * CROW + (16 + tc)] = acc11[rr];
    float v01 = acc01[rr];                    // = C[row][16+tc]
    C[row * CROW + (16 + tc)] = v01;
    C[(16 + tc) * CROW + row] = v01;          // mirror (C symmetric)
  }
  __syncthreads();

  // ---------------- Householder tridiagonalization (lane = column) ----------------
  for (int k = 0; k < A_ATOMS - 2; ++k) {
    float xj  = (lane > k) ? C[lane * CROW + k] : 0.f;   // column k below diagonal
    float sig = wsum32(xj * xj);                          // ||x||^2 (uniform)
    float x0  = __shfl(xj, k + 1, 32);
    float nrm = sqrtf(sig);
    float alpha = 0.f;
    if (nrm >= 1e-18f) {
      alpha = (x0 >= 0.f) ? -nrm : nrm;
      float vj = xj - ((lane == k + 1) ? alpha : 0.f);
      float vn2 = wsum32(vj * vj);
      vj *= rsqrtf(fmaxf(vn2, 1e-30f));                   // normalized Householder vector
      vbuf[lane] = vj;
      __syncthreads();
      float wj = 0.f;                                     // w = B v  (B = trailing block)
      if (lane > k) {
        for (int i = k + 1; i < 32; ++i) wj += C[i * CROW + lane] * vbuf[i];
      }
      float beta = wsum32(vj * wj);                       // v.w (uniform)
      wbuf[lane] = wj;
      __syncthreads();
      if (lane > k) {                                     // B -= 2 v w^T + 2 w v^T - 4*beta v v^T
        float c1 = 2.f * wj, c2 = 2.f * vj, c3 = 4.f * beta * vj;
        for (int i = k + 1; i < 32; ++i) {
          float vi = vbuf[i], wi = wbuf[i];
          C[i * CROW + lane] += vi * c3 - vi * c1 - wi * c2;
        }
      }
      __syncthreads();
    }
    if (lane == 0) ebuf[k] = alpha;                       // subdiagonal entry
    __syncthreads();
  }
  dbuf[lane] = C[lane * CROW + lane];
  if (lane == 0) { ebuf[30] = C[31 * CROW + 30]; ebuf[31] = 0.f; }
  __syncthreads();

  // ---------------- smallest eigenvalue: 32-way Sturm multisection ----------------
  float ei  = ebuf[lane];
  float eim = (lane > 0) ? ebuf[lane - 1] : 0.f;
  float dl  = dbuf[lane];
  float lo  = wmin32(dl - (fabsf(ei) + fabsf(eim)));      // Gershgorin lower bound
  float hi  = wmin32(dl);                                 // lambda_min <= min diag
  lo -= 1e-3f + 1e-6f * fabsf(lo);
  hi += 1e-3f + 1e-6f * fabsf(hi);

#pragma unroll 1
  for (int round = 0; round < 6; ++round) {               // interval shrinks 33x per round
    float x = lo + (hi - lo) * ((float)(lane + 1) * (1.f / 33.f));
    int cnt = 0;                                          // #eigenvalues < x (Sturm)
    float q = dbuf[0] - x;
    if (fabsf(q) < 1e-18f) q = -1e-18f;
    cnt += (q < 0.f);
    for (int i = 1; i < 32; ++i) {
      float e2 = ebuf[i - 1]; e2 *= e2;
      q = (dbuf[i] - x) - e2 / q;
      if (fabsf(q) < 1e-18f) q = -1e-18f;
      cnt += (q < 0.f);
    }
    unsigned mp = (unsigned)__ballot(cnt > 0);            // wave32: low 32 bits
    unsigned mz = ~mp;
    if (mp) hi = __shfl(x, __builtin_ctz(mp), 32);        // smallest shift above lambda_min
    if (mz) lo = __shfl(x, 31 - __builtin_clz(mz), 32);   // largest shift below
  }
  if (lane == 0) out[mol] = 0.5f * (lo + hi);
}

extern "C" void kernel_launch(void* const* d_in, const int* in_sizes, int n_in,
                              void* d_out, int out_size, void* d_ws, size_t ws_size,
                              hipStream_t stream) {
  const float* sr = (const float*)d_in[0];
  // d_in[1] (idx_m) encodes uniform groups of A_ATOMS; layout is implied.
  float* out = (float*)d_out;
  int M = out_size;                       // 8192 molecules
  Correlation_85134841741354_kernel<<<M, 32, 0, stream>>>(sr, out, M);
}